// Conv3d_71339406787348
// MI455X (gfx1250) — compile-verified
//
#include <hip/hip_runtime.h>

typedef float v2f __attribute__((ext_vector_type(2)));
typedef float v8f __attribute__((ext_vector_type(8)));

#define KK   27
#define CIN  16
#define COUT 16
#define WTOT (KK * CIN * COUT)   /* 6912 floats = 27.6 KB */

// One wave computes a 16-node x 16-channel output tile with
// V_WMMA_F32_16X16X4_F32, accumulating over 27 kernel taps x 16 input
// channels (K-dim 432, 108 WMMAs/tile).
__global__ __launch_bounds__(256) void octree_conv_wmma(
    const float* __restrict__ x,      // [N_in, 16]
    const float* __restrict__ w,      // [27, 16, 16] (k, c_in, c_out)
    const float* __restrict__ bias,   // [16]
    const int*   __restrict__ nidx,   // [N_out, 27]
    float* __restrict__ out,          // [N_out, 16]
    int n_out, int n_tiles)
{
    __shared__ __align__(16) float lds_w[WTOT];

    // Cooperative reorder of weights into per-(k,c4) WMMA B-tile layout.
    // Block kc = k*4+c4 holds 64 floats; lane L owns the pair
    //   ( W[k][c4*4 + (L>>4)*2 + 0][L&15], W[k][c4*4 + (L>>4)*2 + 1][L&15] )
    // at lds_w[kc*64 + L*2 + {0,1}], matching the f32 4x16 B VGPR layout
    // (lanes 0-15: K rows {0,1}; lanes 16-31: K rows {2,3}).
    for (int i = threadIdx.x; i < WTOT; i += blockDim.x) {
        int kc  = i >> 6;            // 0..107
        int r   = i & 63;
        int L   = r >> 1;            // lane 0..31
        int p   = r & 1;
        int k   = kc >> 2;
        int c4  = kc & 3;
        int row = c4 * 4 + ((L >> 4) << 1) + p;
        int col = L & 15;
        lds_w[i] = w[(k * CIN + row) * COUT + col];
    }
    __syncthreads();

    const int lane = threadIdx.x & 31;
    const int hi   = lane >> 4;      // selects K pair {0,1} vs {2,3} for A/B
    const int lo   = lane & 15;      // M row (A) / N column (B, D)

    const int wave  = threadIdx.x >> 5;
    const int wpb   = blockDim.x >> 5;
    const int gwave = blockIdx.x * wpb + wave;
    const int nwav  = gridDim.x * wpb;

    const float bv = bias[lo];

    for (int tile = gwave; tile < n_tiles; tile += nwav) {
        int node = tile * 16 + lo;
        if (node >= n_out) node = n_out - 1;   // clamp loads for ragged tail

        // Hoist the 27 neighbor indices of this lane's node into registers.
        int nb[KK];
        #pragma unroll
        for (int k = 0; k < KK; ++k) nb[k] = nidx[node * KK + k];

        v8f acc;
        #pragma unroll
        for (int r = 0; r < 8; ++r) acc[r] = bv;   // D starts at bias

        #pragma unroll
        for (int k = 0; k < KK; ++k) {
            // A row for this lane: gathered input row, channel pair base.
            const float* xr = x + (size_t)nb[k] * CIN + hi * 2;
            #pragma unroll
            for (int c4 = 0; c4 < 4; ++c4) {
                v2f a = *(const v2f*)(xr + c4 * 4);                       // global_load_b64
                v2f b = *(const v2f*)(lds_w + ((k * 4 + c4) << 6) + lane * 2); // ds_load_b64
                acc = __builtin_amdgcn_wmma_f32_16x16x4_f32(
                    false, a, false, b, (short)0, acc, false, false);
            }
        }

        // D layout: VGPR r -> M = r + hi*8, N = lo.
        #pragma unroll
        for (int r = 0; r < 8; ++r) {
            int m = tile * 16 + hi * 8 + r;
            if (m < n_out) out[(size_t)m * COUT + lo] = acc[r];
        }
    }
}

// Reference returns (out, level): append level as the last output element.
__global__ void write_level_tail(const int* __restrict__ level,
                                 float* __restrict__ out, long long pos)
{
    out[pos] = (float)level[0];
}

extern "C" void kernel_launch(void* const* d_in, const int* in_sizes, int n_in,
                              void* d_out, int out_size, void* d_ws, size_t ws_size,
                              hipStream_t stream)
{
    (void)d_ws; (void)ws_size; (void)n_in;

    const float* x     = (const float*)d_in[0];
    const float* w     = (const float*)d_in[1];
    const float* bias  = (const float*)d_in[2];
    const int*   nidx  = (const int*)d_in[3];
    const int*   level = (const int*)d_in[4];
    float*       out   = (float*)d_out;

    const int n_out   = in_sizes[3] / KK;          // neighbor_idx is [N_out, 27]
    const int n_tiles = (n_out + 15) / 16;

    const int threads = 256;                       // 8 waves/block
    int blocks = (n_tiles + 7) / 8;
    if (blocks > 2048) blocks = 2048;              // grid-stride over tiles
    if (blocks < 1) blocks = 1;

    octree_conv_wmma<<<blocks, threads, 0, stream>>>(x, w, bias, nidx, out,
                                                     n_out, n_tiles);

    const long long conv_elems = (long long)n_out * COUT;
    if ((long long)out_size > conv_elems)
        write_level_tail<<<1, 1, 0, stream>>>(level, out, conv_elems);
}